// DGCNN_91336774517538
// MI455X (gfx1250) — compile-verified
//
#include <hip/hip_runtime.h>

// ---------------------------------------------------------------------------
// CDNA5 (gfx1250) fp32 WMMA DGCNN pipeline.
// All matmuls use V_WMMA_F32_16X16X4_F32 (wave32, 16x16 tile, K=4/instr).
// ---------------------------------------------------------------------------

typedef __attribute__((ext_vector_type(2))) float v2f;
typedef __attribute__((ext_vector_type(8))) float v8f;

#define NPTS 16384
#define KNN  20
#define CO   64
#define NEG_HUGE (-3.402823466e38f)
#define POS_HUGE ( 3.402823466e38f)

__device__ __forceinline__ v8f wmma4(v2f a, v2f b, v8f c) {
  // (neg_a, A, neg_b, B, c_mod, C, reuse_a, reuse_b)
  return __builtin_amdgcn_wmma_f32_16x16x4_f32(false, a, false, b, (short)0, c,
                                               false, false);
}

__device__ __forceinline__ v8f zero8() {
  v8f z = {0.f, 0.f, 0.f, 0.f, 0.f, 0.f, 0.f, 0.f};
  return z;
}

__device__ __forceinline__ float prelu1(float v, float a) {
  return fmaxf(v, 0.f) + a * fminf(v, 0.f);
}

// Sorted ascending top-K insertion, all register indices static (no spills).
__device__ __forceinline__ void topk_insert(float d, int idx, float dk[KNN], int ik[KNN]) {
  if (d < dk[KNN - 1]) {
#pragma unroll
    for (int s = KNN - 1; s >= 1; --s) {
      const float lo = dk[s - 1];
      const int   loi = ik[s - 1];
      if (d < lo)          { dk[s] = lo; ik[s] = loi; }
      else if (d < dk[s])  { dk[s] = d;  ik[s] = idx; }
    }
    if (d < dk[0]) { dk[0] = d; ik[0] = idx; }
  }
}

__device__ __forceinline__ void atomic_max_float(float* addr, float val) {
  if (val >= 0.f) atomicMax((int*)addr, __float_as_int(val));
  else            atomicMin((unsigned int*)addr, __float_as_uint(val));
}

// ---------------------------------------------------------------------------
// 1) Brute-force knn on [N,3] points (too skinny for WMMA). LDS-tiled.
// ---------------------------------------------------------------------------
__global__ __launch_bounds__(256) void knn_c3_kernel(const float* __restrict__ x,
                                                     int* __restrict__ ind) {
  __shared__ float xs[1024 * 3];
  const int q = blockIdx.x * 256 + threadIdx.x;
  const float qx = x[q * 3 + 0], qy = x[q * 3 + 1], qz = x[q * 3 + 2];
  float dk[KNN]; int ik[KNN];
#pragma unroll
  for (int t = 0; t < KNN; ++t) { dk[t] = POS_HUGE; ik[t] = 0; }
#pragma unroll 1
  for (int c0 = 0; c0 < NPTS; c0 += 1024) {
    __syncthreads();
    for (int i = threadIdx.x; i < 1024 * 3; i += 256) xs[i] = x[c0 * 3 + i];
    __syncthreads();
#pragma unroll 1
    for (int j = 0; j < 1024; ++j) {
      const float dx = xs[j * 3 + 0] - qx;
      const float dy = xs[j * 3 + 1] - qy;
      const float dz = xs[j * 3 + 2] - qz;
      topk_insert(dx * dx + dy * dy + dz * dz, c0 + j, dk, ik);
    }
  }
#pragma unroll
  for (int t = 0; t < KNN; ++t) ind[q * KNN + t] = ik[t];
}

// ---------------------------------------------------------------------------
// 2) Squared norms of 64-dim features.
// ---------------------------------------------------------------------------
__global__ __launch_bounds__(256) void norm64_kernel(const float* __restrict__ xf,
                                                     float* __restrict__ nrm) {
  const int i = blockIdx.x * 256 + threadIdx.x;
  float s = 0.f;
#pragma unroll
  for (int c = 0; c < CO; ++c) { const float v = xf[i * CO + c]; s += v * v; }
  nrm[i] = s;
}

// ---------------------------------------------------------------------------
// 3) knn on [N,64] features: Gram tiles via fp32 WMMA, fused top-k.
//    Block = 8 waves; each wave owns 16 query rows (A fragments preloaded).
// ---------------------------------------------------------------------------
__global__ __launch_bounds__(256) void knn_feat_wmma_kernel(const float* __restrict__ xf,
                                                            const float* __restrict__ nrm,
                                                            int* __restrict__ ind) {
  __shared__ float cand[16 * CO];     // shared candidate tile (all 8 waves)
  __shared__ float dt[8][16 * 16];    // per-wave distance tile
  const int lane = threadIdx.x & 31;
  const int wv   = threadIdx.x >> 5;
  const int q0   = blockIdx.x * 128 + wv * 16;
  const int row  = lane & 15;         // A: query row; B: candidate col
  const int half = lane >> 4;

  // A fragments: 16 queries x 64 dims, resident in registers for whole kernel.
  v2f af[16];
#pragma unroll
  for (int kk = 0; kk < 16; ++kk) {
    const int ks = 4 * kk + 2 * half;
    af[kk].x = xf[(q0 + row) * CO + ks];
    af[kk].y = xf[(q0 + row) * CO + ks + 1];
  }
  float qn[8];
#pragma unroll
  for (int r = 0; r < 8; ++r) qn[r] = nrm[q0 + r + 8 * half];

  float dk[KNN]; int ik[KNN];
#pragma unroll
  for (int t = 0; t < KNN; ++t) { dk[t] = POS_HUGE; ik[t] = 0; }

#pragma unroll 1
  for (int c0 = 0; c0 < NPTS; c0 += 16) {
    __syncthreads();
    for (int i = threadIdx.x; i < 16 * CO; i += 256) cand[i] = xf[c0 * CO + i];
    __syncthreads();

    v8f acc = zero8();
#pragma unroll
    for (int kk = 0; kk < 16; ++kk) {
      const int ks = 4 * kk + 2 * half;
      v2f b; b.x = cand[row * CO + ks]; b.y = cand[row * CO + ks + 1];
      acc = wmma4(af[kk], b, acc);
    }
    const float cn = nrm[c0 + row];
#pragma unroll
    for (int r = 0; r < 8; ++r)
      dt[wv][(r + 8 * half) * 16 + row] = qn[r] + cn - 2.f * acc[r];
    __syncthreads();
    if (lane < 16) {
#pragma unroll 1
      for (int j = 0; j < 16; ++j)
        topk_insert(dt[wv][lane * 16 + j], c0 + j, dk, ik);
    }
  }
  if (lane < 16) {
#pragma unroll
    for (int t = 0; t < KNN; ++t) ind[(q0 + lane) * KNN + t] = ik[t];
  }
}

// ---------------------------------------------------------------------------
// 4) Fused edge-conv: gather -> MLP (1 or 2 WMMA layers) -> max over K nbrs.
//    Wave = 16 points; feature rows synthesized directly into A fragments.
// ---------------------------------------------------------------------------
__device__ __forceinline__ float edge_feat(const float* __restrict__ x, int p, int j,
                                           int cin, int cin2, int e) {
  if (e >= cin2) return 0.f;
  if (e < cin)  return x[j * cin + e] - x[p * cin + e];
  return x[p * cin + (e - cin)];
}

__global__ __launch_bounds__(256) void edge_conv_kernel(
    const float* __restrict__ x, int cin,
    const int* __restrict__ ind,
    const float* __restrict__ Wa, const float* __restrict__ pa,
    const float* __restrict__ Wb, const float* __restrict__ pb,
    float* __restrict__ out) {
  __shared__ float ya[8][16 * CO];    // per-wave layer-a output (transpose hop)
  const int lane = threadIdx.x & 31;
  const int wv   = threadIdx.x >> 5;
  const int q0   = blockIdx.x * 128 + wv * 16;
  const int row  = lane & 15;
  const int half = lane >> 4;
  const int col  = lane & 15;
  const int p    = q0 + row;
  const int cin2 = 2 * cin;
  const int nchunk = (cin2 + 3) >> 2;
  const float aa = pa[0];
  const bool twolayer = (Wb != nullptr);
  const float ab = twolayer ? pb[0] : 0.f;

  v8f accmax[4];
#pragma unroll
  for (int nt = 0; nt < 4; ++nt) accmax[nt] = zero8();
#pragma unroll
  for (int nt = 0; nt < 4; ++nt)
#pragma unroll
    for (int r = 0; r < 8; ++r) accmax[nt][r] = NEG_HUGE;

#pragma unroll 1
  for (int t = 0; t < KNN; ++t) {
    const int j = ind[p * KNN + t];
    v8f acc[4];
#pragma unroll
    for (int nt = 0; nt < 4; ++nt) acc[nt] = zero8();

    // layer a: [16 x cin2] @ Wa[cin2 x 64]
#pragma unroll 1
    for (int kk = 0; kk < nchunk; ++kk) {
      const int ks = 4 * kk + 2 * half;
      v2f a;
      a.x = edge_feat(x, p, j, cin, cin2, ks);
      a.y = edge_feat(x, p, j, cin, cin2, ks + 1);
#pragma unroll
      for (int nt = 0; nt < 4; ++nt) {
        const int c = nt * 16 + col;
        v2f b;
        b.x = (ks     < cin2) ? Wa[ks       * CO + c] : 0.f;
        b.y = (ks + 1 < cin2) ? Wa[(ks + 1) * CO + c] : 0.f;
        acc[nt] = wmma4(a, b, acc[nt]);
      }
    }
#pragma unroll
    for (int nt = 0; nt < 4; ++nt)
#pragma unroll
      for (int r = 0; r < 8; ++r) acc[nt][r] = prelu1(acc[nt][r], aa);

    if (twolayer) {   // layer b: [16 x 64] @ Wb[64 x 64], transpose via LDS
#pragma unroll
      for (int nt = 0; nt < 4; ++nt)
#pragma unroll
        for (int r = 0; r < 8; ++r)
          ya[wv][(r + 8 * half) * CO + nt * 16 + col] = acc[nt][r];
      __syncthreads();
      v8f acc2[4];
#pragma unroll
      for (int nt = 0; nt < 4; ++nt) acc2[nt] = zero8();
#pragma unroll 1
      for (int kk = 0; kk < 16; ++kk) {
        const int ks = 4 * kk + 2 * half;
        v2f a2; a2.x = ya[wv][row * CO + ks]; a2.y = ya[wv][row * CO + ks + 1];
#pragma unroll
        for (int nt = 0; nt < 4; ++nt) {
          const int c = nt * 16 + col;
          v2f b; b.x = Wb[ks * CO + c]; b.y = Wb[(ks + 1) * CO + c];
          acc2[nt] = wmma4(a2, b, acc2[nt]);
        }
      }
      __syncthreads();
#pragma unroll
      for (int nt = 0; nt < 4; ++nt)
#pragma unroll
        for (int r = 0; r < 8; ++r) acc[nt][r] = prelu1(acc2[nt][r], ab);
    }

#pragma unroll
    for (int nt = 0; nt < 4; ++nt)
#pragma unroll
      for (int r = 0; r < 8; ++r) accmax[nt][r] = fmaxf(accmax[nt][r], acc[nt][r]);
  }

#pragma unroll
  for (int nt = 0; nt < 4; ++nt)
#pragma unroll
    for (int r = 0; r < 8; ++r)
      out[(q0 + r + 8 * half) * CO + nt * 16 + col] = accmax[nt][r];
}

// ---------------------------------------------------------------------------
// 5) Dense WMMA GEMM + PReLU (+optional bias). Wave owns a 16x32 output tile.
// ---------------------------------------------------------------------------
__global__ __launch_bounds__(256) void gemm_prelu_kernel(
    const float* __restrict__ A, int Kd,
    const float* __restrict__ W, int Nd,
    const float* __restrict__ bias,
    const float* __restrict__ alpha_p,
    float* __restrict__ out) {
  const int lane = threadIdx.x & 31;
  const int wv   = threadIdx.x >> 5;
  const int m0   = (blockIdx.y * 8 + wv) * 16;
  const int n0   = blockIdx.x * 32;
  const int row  = lane & 15;
  const int half = lane >> 4;
  const int col  = lane & 15;
  const float alpha = alpha_p[0];
  const int nc0 = n0 + col, nc1 = n0 + 16 + col;
  v8f acc0 = zero8(), acc1 = zero8();
#pragma unroll 1
  for (int kk = 0; kk < (Kd >> 2); ++kk) {
    const int ks = 4 * kk + 2 * half;
    v2f a; a.x = A[(m0 + row) * Kd + ks]; a.y = A[(m0 + row) * Kd + ks + 1];
    v2f b0, b1;
    b0.x = (nc0 < Nd) ? W[ks * Nd + nc0] : 0.f;
    b0.y = (nc0 < Nd) ? W[(ks + 1) * Nd + nc0] : 0.f;
    b1.x = (nc1 < Nd) ? W[ks * Nd + nc1] : 0.f;
    b1.y = (nc1 < Nd) ? W[(ks + 1) * Nd + nc1] : 0.f;
    acc0 = wmma4(a, b0, acc0);
    acc1 = wmma4(a, b1, acc1);
  }
#pragma unroll
  for (int r = 0; r < 8; ++r) {
    const int m = m0 + r + 8 * half;
    if (nc0 < Nd) {
      float v = acc0[r]; if (bias) v += bias[nc0];
      out[m * Nd + nc0] = prelu1(v, alpha);
    }
    if (nc1 < Nd) {
      float v = acc1[r]; if (bias) v += bias[nc1];
      out[m * Nd + nc1] = prelu1(v, alpha);
    }
  }
}

// ---------------------------------------------------------------------------
// 6) GEMM over virtual concat [x1|x2|x3] (K=192). colmax_mode: PReLU +
//    atomic column-max epilogue (for the global max-pool feature).
// ---------------------------------------------------------------------------
__device__ __forceinline__ float cat3(const float* __restrict__ x1,
                                      const float* __restrict__ x2,
                                      const float* __restrict__ x3,
                                      int r, int k) {
  const float* s = (k < 64) ? x1 : ((k < 128) ? x2 : x3);
  return s[r * CO + (k & 63)];
}

__global__ __launch_bounds__(256) void gemm_cat3_kernel(
    const float* __restrict__ x1, const float* __restrict__ x2,
    const float* __restrict__ x3,
    const float* __restrict__ W, int Nd,
    const float* __restrict__ bias,
    const float* __restrict__ alpha_p,
    float* __restrict__ out, int colmax_mode) {
  const int lane = threadIdx.x & 31;
  const int wv   = threadIdx.x >> 5;
  const int m0   = (blockIdx.y * 8 + wv) * 16;
  const int n0   = blockIdx.x * 32;
  const int row  = lane & 15;
  const int half = lane >> 4;
  const int col  = lane & 15;
  const float alpha = alpha_p[0];
  const int nc0 = n0 + col, nc1 = n0 + 16 + col;
  v8f acc0 = zero8(), acc1 = zero8();
#pragma unroll 1
  for (int kk = 0; kk < 48; ++kk) {          // K = 192
    const int ks = 4 * kk + 2 * half;
    v2f a;
    a.x = cat3(x1, x2, x3, m0 + row, ks);
    a.y = cat3(x1, x2, x3, m0 + row, ks + 1);
    v2f b0, b1;
    b0.x = (nc0 < Nd) ? W[ks * Nd + nc0] : 0.f;
    b0.y = (nc0 < Nd) ? W[(ks + 1) * Nd + nc0] : 0.f;
    b1.x = (nc1 < Nd) ? W[ks * Nd + nc1] : 0.f;
    b1.y = (nc1 < Nd) ? W[(ks + 1) * Nd + nc1] : 0.f;
    acc0 = wmma4(a, b0, acc0);
    acc1 = wmma4(a, b1, acc1);
  }
  if (colmax_mode) {
    float mx0 = NEG_HUGE, mx1 = NEG_HUGE;
#pragma unroll
    for (int r = 0; r < 8; ++r) {
      mx0 = fmaxf(mx0, prelu1(acc0[r], alpha));
      mx1 = fmaxf(mx1, prelu1(acc1[r], alpha));
    }
    if (nc0 < Nd) atomic_max_float(out + nc0, mx0);
    if (nc1 < Nd) atomic_max_float(out + nc1, mx1);
  } else {
#pragma unroll
    for (int r = 0; r < 8; ++r) {
      const int m = m0 + r + 8 * half;
      if (nc0 < Nd) {
        float v = acc0[r]; if (bias) v += bias[nc0];
        out[m * Nd + nc0] = prelu1(v, alpha);
      }
      if (nc1 < Nd) {
        float v = acc1[r]; if (bias) v += bias[nc1];
        out[m * Nd + nc1] = prelu1(v, alpha);
      }
    }
  }
}

// ---------------------------------------------------------------------------
// Small helpers.
// ---------------------------------------------------------------------------
__global__ void init_colmax_kernel(float* __restrict__ cm) {
  const int i = blockIdx.x * 256 + threadIdx.x;
  if (i < 1024) cm[i] = NEG_HUGE;
}

// bias[j] = colmax[0:1024] . W5[192:1216, j]  (broadcast-row contribution)
__global__ void bias5_kernel(const float* __restrict__ cm,
                             const float* __restrict__ W5,
                             float* __restrict__ bias) {
  const int j = threadIdx.x;  // 256 threads
  float s = 0.f;
#pragma unroll 1
  for (int k = 0; k < 1024; ++k) s += cm[k] * W5[(192 + k) * 256 + j];
  bias[j] = s;
}

__global__ void tail_kernel(const float* __restrict__ hidden, float* __restrict__ out) {
  out[NPTS * 3] = hidden[0];
}

// ---------------------------------------------------------------------------
// Launch sequence.
// ---------------------------------------------------------------------------
extern "C" void kernel_launch(void* const* d_in, const int* in_sizes, int n_in,
                              void* d_out, int out_size, void* d_ws, size_t ws_size,
                              hipStream_t stream) {
  (void)in_sizes; (void)n_in; (void)out_size; (void)ws_size;
  const float* x   = (const float*)d_in[0];
  const float* hid = (const float*)d_in[1];
  const float* W1a = (const float*)d_in[2];  const float* p1a = (const float*)d_in[3];
  const float* W1b = (const float*)d_in[4];  const float* p1b = (const float*)d_in[5];
  const float* W2a = (const float*)d_in[6];  const float* p2a = (const float*)d_in[7];
  const float* W2b = (const float*)d_in[8];  const float* p2b = (const float*)d_in[9];
  const float* W3  = (const float*)d_in[10]; const float* p3  = (const float*)d_in[11];
  const float* W4  = (const float*)d_in[12]; const float* p4  = (const float*)d_in[13];
  const float* W5  = (const float*)d_in[14]; const float* p5  = (const float*)d_in[15];
  const float* W6  = (const float*)d_in[16]; const float* p6  = (const float*)d_in[17];
  const float* W7  = (const float*)d_in[18]; const float* p7  = (const float*)d_in[19];
  const float* W8  = (const float*)d_in[20]; const float* p8  = (const float*)d_in[21];

  // workspace layout (~54 MB total)
  char* ws = (char*)d_ws;
  size_t off = 0;
  auto take = [&](size_t bytes) -> void* {
    void* p = ws + off;
    off = (off + bytes + 255) & ~(size_t)255;
    return p;
  };
  int*   ind    = (int*)  take((size_t)NPTS * KNN * 4);
  float* norms  = (float*)take((size_t)NPTS * 4);
  float* x1     = (float*)take((size_t)NPTS * CO * 4);
  float* x2     = (float*)take((size_t)NPTS * CO * 4);
  float* x3     = (float*)take((size_t)NPTS * CO * 4);
  float* colmax = (float*)take(1024 * 4);
  float* bias5  = (float*)take(256 * 4);
  float* h5     = (float*)take((size_t)NPTS * 256 * 4);
  float* h6     = (float*)take((size_t)NPTS * 256 * 4);
  float* h7     = (float*)take((size_t)NPTS * 128 * 4);

  // edge conv 1 (C=3)
  knn_c3_kernel<<<NPTS / 256, 256, 0, stream>>>(x, ind);
  edge_conv_kernel<<<NPTS / 128, 256, 0, stream>>>(x, 3, ind, W1a, p1a, W1b, p1b, x1);

  // edge conv 2 (C=64)
  norm64_kernel<<<NPTS / 256, 256, 0, stream>>>(x1, norms);
  knn_feat_wmma_kernel<<<NPTS / 128, 256, 0, stream>>>(x1, norms, ind);
  edge_conv_kernel<<<NPTS / 128, 256, 0, stream>>>(x1, 64, ind, W2a, p2a, W2b, p2b, x2);

  // edge conv 3 (C=64, single layer)
  norm64_kernel<<<NPTS / 256, 256, 0, stream>>>(x2, norms);
  knn_feat_wmma_kernel<<<NPTS / 128, 256, 0, stream>>>(x2, norms, ind);
  edge_conv_kernel<<<NPTS / 128, 256, 0, stream>>>(x2, 64, ind, W3, p3, nullptr, nullptr, x3);

  // global feature: colmax = max_rows prelu([x1|x2|x3] @ W4)
  init_colmax_kernel<<<4, 256, 0, stream>>>(colmax);
  gemm_cat3_kernel<<<dim3(1024 / 32, NPTS / 128), 256, 0, stream>>>(
      x1, x2, x3, W4, 1024, nullptr, p4, colmax, 1);
  bias5_kernel<<<1, 256, 0, stream>>>(colmax, W5, bias5);

  // head: h5 = prelu([x1|x2|x3] @ W5[:192] + colmax @ W5[192:])
  gemm_cat3_kernel<<<dim3(256 / 32, NPTS / 128), 256, 0, stream>>>(
      x1, x2, x3, W5, 256, bias5, p5, h5, 0);
  gemm_prelu_kernel<<<dim3(256 / 32, NPTS / 128), 256, 0, stream>>>(h5, 256, W6, 256, nullptr, p6, h6);
  gemm_prelu_kernel<<<dim3(128 / 32, NPTS / 128), 256, 0, stream>>>(h6, 256, W7, 128, nullptr, p7, h7);
  gemm_prelu_kernel<<<dim3(1, NPTS / 128), 256, 0, stream>>>(h7, 128, W8, 3, nullptr, p8, (float*)d_out);

  tail_kernel<<<1, 1, 0, stream>>>(hid, (float*)d_out);
}